// Expert_67181878444420
// MI455X (gfx1250) — compile-verified
//
#include <hip/hip_runtime.h>
#include <math.h>
#include <stdint.h>

typedef __attribute__((ext_vector_type(2))) float v2f;
typedef __attribute__((ext_vector_type(8))) float v8f;

#define WAVES 8

__device__ __forceinline__ v8f wmma4(v2f a, v2f b, v8f c) {
  // D = A(16x4,f32) * B(4x16,f32) + C(16x16,f32)
  return __builtin_amdgcn_wmma_f32_16x16x4_f32(
      /*neg_a=*/false, a, /*neg_b=*/false, b,
      /*c_mod=*/(short)0, c, /*reuse_a=*/false, /*reuse_b=*/false);
}

// sigmoid via v_exp_f32 + v_rcp_f32 (no IEEE divide sequence)
__device__ __forceinline__ float fast_sigmoid(float x) {
  return __builtin_amdgcn_rcpf(1.0f + __expf(-x));
}

#if __has_builtin(__builtin_amdgcn_global_load_async_to_lds_b128) && \
    __has_builtin(__builtin_amdgcn_s_wait_asynccnt)
#define USE_ASYNC_LDS 1
// Probe-learned parameter type: generic pointer to int __vector_size(16).
typedef int vint4_t __attribute__((vector_size(16)));
#endif

// B fragment: rows ka,ka+1 of W[K][C] at column col.
// PACKED layout: Wp[((k>>1)*C + col)*2 + (k&1)]  -> one global_load_b64.
template <bool PACKED>
__device__ __forceinline__ v2f loadB(const float* __restrict__ W, int C,
                                     int ka, int col) {
  v2f b;
  if constexpr (PACKED) {
    b = *(const v2f*)(W + ((((size_t)(ka >> 1)) * C + col) << 1));
  } else {
    b.x = W[(size_t)ka * C + col];
    b.y = W[(size_t)(ka + 1) * C + col];
  }
  return b;
}

// ---- 32-node tile: two 16-row WMMA A-tiles share every B fragment ----
#define XL_STRIDE 706
#define S_STRIDE  674
#define X0_STRIDE 258
#define Z1_STRIDE 194
#define Z2_STRIDE  98
#define OFF_XL   0
#define OFF_S    (OFF_XL + 32 * XL_STRIDE)                  // 22592
#define OFF_U    (OFF_S + 32 * S_STRIDE)                    // 44160 (union)
#define OFF_X0   OFF_U
#define OFF_Z1   OFF_U
#define OFF_Z2   (OFF_U + 3 * 32 * Z1_STRIDE)
#define OFF_ATTR (OFF_U + 3 * 32 * Z1_STRIDE + 5 * 32 * Z2_STRIDE)  // 78464
#define SMEM_DW  (OFF_ATTR + 32)                            // 313984 B

template <bool PACKED>
__global__ __launch_bounds__(256)
void fused_gated_mlp(const float* __restrict__ node_input,
                     const float* __restrict__ node_attr,
                     const float* __restrict__ w1_s,
                     const float* __restrict__ b1_s,
                     const float* __restrict__ w1_l1,
                     const float* __restrict__ w1_l2,
                     const float* __restrict__ w2_s,
                     const float* __restrict__ b2_s,
                     const float* __restrict__ w2_l1,
                     const float* __restrict__ w2_l2,
                     float* __restrict__ out) {
  __shared__ float smem[SMEM_DW];

  const int tid  = threadIdx.x;
  const int wave = __builtin_amdgcn_readfirstlane(tid >> 5);  // scalar wave id
  const int lane = tid & 31;
  const int l16  = lane & 15;
  const int hi   = lane >> 4;
  const int node0 = blockIdx.x * 32;

  if (tid < 6) {
    const void* wp[6] = {w1_s, w1_l1, w1_l2, w2_s, w2_l1, w2_l2};
    __builtin_prefetch(wp[tid], 0, 2);
  }

  // ---- Stage 0: async-DMA the 32x960 node tile into LDS ----
  for (int idx = tid; idx < 32 * 240; idx += 256) {
    const int r  = idx / 240;
    const int c4 = idx % 240;
    const float* src = node_input + (size_t)(node0 + r) * 960 + c4 * 4;
    float* dst = (c4 < 64) ? &smem[OFF_X0 + r * X0_STRIDE + c4 * 4]
                           : &smem[OFF_XL + r * XL_STRIDE + (c4 - 64) * 4];
#if defined(USE_ASYNC_LDS)
    __builtin_amdgcn_global_load_async_to_lds_b128(
        (vint4_t*)src, (vint4_t*)dst, 0, 0);
#else
    *(float4*)dst = *(const float4*)src;
#endif
  }
  if (tid < 32) smem[OFF_ATTR + tid] = node_attr[node0 + tid];
#if defined(USE_ASYNC_LDS)
  __builtin_amdgcn_s_wait_asynccnt(0);
#endif
  __syncthreads();

  float attr0[8], attr1[8];
#pragma unroll
  for (int v = 0; v < 8; ++v) {
    attr0[v] = smem[OFF_ATTR + v + 8 * hi];
    attr1[v] = smem[OFF_ATTR + 16 + v + 8 * hi];
  }

  // ---- Stage 1s: s = act( x0 @ w1_s * (a/16) + b1 ), 42 column tiles ----
  for (int nt = wave; nt < 42; nt += WAVES) {
    const int colb = nt * 16;
    const int col  = colb + l16;
    v8f acc0 = {}, acc1 = {}, acc2 = {}, acc3 = {};
    const float* arow0 = &smem[OFF_X0 + l16 * X0_STRIDE];
    const float* arow1 = &smem[OFF_X0 + (16 + l16) * X0_STRIDE];
#pragma unroll 2
    for (int kk = 0; kk < 64; kk += 2) {
      const int ka = 4 * kk + 2 * hi;
      const int kb = ka + 4;
      v2f a00 = *(const v2f*)(arow0 + ka);
      v2f a01 = *(const v2f*)(arow0 + kb);
      v2f a10 = *(const v2f*)(arow1 + ka);
      v2f a11 = *(const v2f*)(arow1 + kb);
      v2f b0 = loadB<PACKED>(w1_s, 672, ka, col);
      v2f b1 = loadB<PACKED>(w1_s, 672, kb, col);
      acc0 = wmma4(a00, b0, acc0);
      acc1 = wmma4(a01, b1, acc1);
      acc2 = wmma4(a10, b0, acc2);
      acc3 = wmma4(a11, b1, acc3);
    }
    const v8f   accA = acc0 + acc1;
    const v8f   accB = acc2 + acc3;
    const float bias = b1_s[col];
    if (colb < 384) {   // uniform (nt is scalar)
#pragma unroll
      for (int v = 0; v < 8; ++v) {
        const float sA = accA[v] * (attr0[v] * 0.0625f) + bias;
        const float sB = accB[v] * (attr1[v] * 0.0625f) + bias;
        smem[OFF_S + (v + 8 * hi) * S_STRIDE + col]      = sA * fast_sigmoid(sA);
        smem[OFF_S + (16 + v + 8 * hi) * S_STRIDE + col] = sB * fast_sigmoid(sB);
      }
    } else {
#pragma unroll
      for (int v = 0; v < 8; ++v) {
        const float sA = accA[v] * (attr0[v] * 0.0625f) + bias;
        const float sB = accB[v] * (attr1[v] * 0.0625f) + bias;
        smem[OFF_S + (v + 8 * hi) * S_STRIDE + col]      = fast_sigmoid(sA);
        smem[OFF_S + (16 + v + 8 * hi) * S_STRIDE + col] = fast_sigmoid(sB);
      }
    }
  }
  __syncthreads();  // g ready; x0 region dead -> reusable as z1/z2

  // ---- Stage 1l1: z1_i = (x1_i @ w1_l1) * (a/sqrt(128)) * g[:192] ----
  for (int w = wave; w < 36; w += WAVES) {
    const int i = w / 12, nt = w % 12;
    const int col = nt * 16 + l16;
    v8f acc0 = {}, acc1 = {}, acc2 = {}, acc3 = {};
    const float* arow0 = &smem[OFF_XL + l16 * XL_STRIDE + i];
    const float* arow1 = &smem[OFF_XL + (16 + l16) * XL_STRIDE + i];
#pragma unroll 2
    for (int kk = 0; kk < 32; kk += 2) {
      const int ka = 4 * kk + 2 * hi;
      const int kb = ka + 4;
      v2f a00, a01, a10, a11;
      a00.x = arow0[3 * ka];      a00.y = arow0[3 * ka + 3];
      a01.x = arow0[3 * kb];      a01.y = arow0[3 * kb + 3];
      a10.x = arow1[3 * ka];      a10.y = arow1[3 * ka + 3];
      a11.x = arow1[3 * kb];      a11.y = arow1[3 * kb + 3];
      v2f b0 = loadB<PACKED>(w1_l1, 192, ka, col);
      v2f b1 = loadB<PACKED>(w1_l1, 192, kb, col);
      acc0 = wmma4(a00, b0, acc0);
      acc1 = wmma4(a01, b1, acc1);
      acc2 = wmma4(a10, b0, acc2);
      acc3 = wmma4(a11, b1, acc3);
    }
    const v8f accA = acc0 + acc1;
    const v8f accB = acc2 + acc3;
#pragma unroll
    for (int v = 0; v < 8; ++v) {
      const int nA = v + 8 * hi, nB = 16 + v + 8 * hi;
      const float gA = smem[OFF_S + nA * S_STRIDE + 384 + col];
      const float gB = smem[OFF_S + nB * S_STRIDE + 384 + col];
      smem[OFF_Z1 + (i * 32 + nA) * Z1_STRIDE + col] =
          accA[v] * (attr0[v] * 0.08838834764831845f) * gA;
      smem[OFF_Z1 + (i * 32 + nB) * Z1_STRIDE + col] =
          accB[v] * (attr1[v] * 0.08838834764831845f) * gB;
    }
  }
  // ---- Stage 1l2: z2_i = (x2_i @ w1_l2) * (a/8) * g[192:288] ----
  for (int w = wave; w < 30; w += WAVES) {
    const int i = w / 6, nt = w % 6;
    const int col = nt * 16 + l16;
    v8f acc0 = {}, acc1 = {}, acc2 = {}, acc3 = {};
    const float* arow0 = &smem[OFF_XL + l16 * XL_STRIDE + 384 + i];
    const float* arow1 = &smem[OFF_XL + (16 + l16) * XL_STRIDE + 384 + i];
#pragma unroll 2
    for (int kk = 0; kk < 16; kk += 2) {
      const int ka = 4 * kk + 2 * hi;
      const int kb = ka + 4;
      v2f a00, a01, a10, a11;
      a00.x = arow0[5 * ka];      a00.y = arow0[5 * ka + 5];
      a01.x = arow0[5 * kb];      a01.y = arow0[5 * kb + 5];
      a10.x = arow1[5 * ka];      a10.y = arow1[5 * ka + 5];
      a11.x = arow1[5 * kb];      a11.y = arow1[5 * kb + 5];
      v2f b0 = loadB<PACKED>(w1_l2, 96, ka, col);
      v2f b1 = loadB<PACKED>(w1_l2, 96, kb, col);
      acc0 = wmma4(a00, b0, acc0);
      acc1 = wmma4(a01, b1, acc1);
      acc2 = wmma4(a10, b0, acc2);
      acc3 = wmma4(a11, b1, acc3);
    }
    const v8f accA = acc0 + acc1;
    const v8f accB = acc2 + acc3;
#pragma unroll
    for (int v = 0; v < 8; ++v) {
      const int nA = v + 8 * hi, nB = 16 + v + 8 * hi;
      const float gA = smem[OFF_S + nA * S_STRIDE + 576 + col];
      const float gB = smem[OFF_S + nB * S_STRIDE + 576 + col];
      smem[OFF_Z2 + (i * 32 + nA) * Z2_STRIDE + col] =
          accA[v] * (attr0[v] * 0.125f) * gA;
      smem[OFF_Z2 + (i * 32 + nB) * Z2_STRIDE + col] =
          accB[v] * (attr1[v] * 0.125f) * gB;
    }
  }
  __syncthreads();  // z1/z2 and sc ready

  // ---- Stage 2s: o0 = sc @ w2_s * (a/sqrt(384)) + b2 ----
  for (int nt = wave; nt < 16; nt += WAVES) {
    const int col = nt * 16 + l16;
    v8f acc0 = {}, acc1 = {}, acc2 = {}, acc3 = {};
    const float* arow0 = &smem[OFF_S + l16 * S_STRIDE];
    const float* arow1 = &smem[OFF_S + (16 + l16) * S_STRIDE];
#pragma unroll 2
    for (int kk = 0; kk < 96; kk += 2) {
      const int ka = 4 * kk + 2 * hi;
      const int kb = ka + 4;
      v2f a00 = *(const v2f*)(arow0 + ka);
      v2f a01 = *(const v2f*)(arow0 + kb);
      v2f a10 = *(const v2f*)(arow1 + ka);
      v2f a11 = *(const v2f*)(arow1 + kb);
      v2f b0 = loadB<PACKED>(w2_s, 256, ka, col);
      v2f b1 = loadB<PACKED>(w2_s, 256, kb, col);
      acc0 = wmma4(a00, b0, acc0);
      acc1 = wmma4(a01, b1, acc1);
      acc2 = wmma4(a10, b0, acc2);
      acc3 = wmma4(a11, b1, acc3);
    }
    const v8f   accA = acc0 + acc1;
    const v8f   accB = acc2 + acc3;
    const float bias = b2_s[col];
#pragma unroll
    for (int v = 0; v < 8; ++v) {
      out[(size_t)(node0 + v + 8 * hi) * 960 + col] =
          accA[v] * (attr0[v] * 0.051031036307982884f) + bias;
      out[(size_t)(node0 + 16 + v + 8 * hi) * 960 + col] =
          accB[v] * (attr1[v] * 0.051031036307982884f) + bias;
    }
  }
  // ---- Stage 2l1: o1_i = z1_i @ w2_l1 * (a/sqrt(192)) ----
  for (int w = wave; w < 24; w += WAVES) {
    const int i = w / 8, nt = w % 8;
    const int col = nt * 16 + l16;
    v8f acc0 = {}, acc1 = {}, acc2 = {}, acc3 = {};
    const float* arow0 = &smem[OFF_Z1 + (i * 32 + l16) * Z1_STRIDE];
    const float* arow1 = &smem[OFF_Z1 + (i * 32 + 16 + l16) * Z1_STRIDE];
#pragma unroll 2
    for (int kk = 0; kk < 48; kk += 2) {
      const int ka = 4 * kk + 2 * hi;
      const int kb = ka + 4;
      v2f a00 = *(const v2f*)(arow0 + ka);
      v2f a01 = *(const v2f*)(arow0 + kb);
      v2f a10 = *(const v2f*)(arow1 + ka);
      v2f a11 = *(const v2f*)(arow1 + kb);
      v2f b0 = loadB<PACKED>(w2_l1, 128, ka, col);
      v2f b1 = loadB<PACKED>(w2_l1, 128, kb, col);
      acc0 = wmma4(a00, b0, acc0);
      acc1 = wmma4(a01, b1, acc1);
      acc2 = wmma4(a10, b0, acc2);
      acc3 = wmma4(a11, b1, acc3);
    }
    const v8f accA = acc0 + acc1;
    const v8f accB = acc2 + acc3;
#pragma unroll
    for (int v = 0; v < 8; ++v) {
      out[(size_t)(node0 + v + 8 * hi) * 960 + 256 + col * 3 + i] =
          accA[v] * (attr0[v] * 0.07216878364870323f);
      out[(size_t)(node0 + 16 + v + 8 * hi) * 960 + 256 + col * 3 + i] =
          accB[v] * (attr1[v] * 0.07216878364870323f);
    }
  }
  // ---- Stage 2l2: o2_i = z2_i @ w2_l2 * (a/sqrt(96)) ----
  for (int w = wave; w < 20; w += WAVES) {
    const int i = w / 4, nt = w % 4;
    const int col = nt * 16 + l16;
    v8f acc0 = {}, acc1 = {}, acc2 = {}, acc3 = {};
    const float* arow0 = &smem[OFF_Z2 + (i * 32 + l16) * Z2_STRIDE];
    const float* arow1 = &smem[OFF_Z2 + (i * 32 + 16 + l16) * Z2_STRIDE];
#pragma unroll 2
    for (int kk = 0; kk < 24; kk += 2) {
      const int ka = 4 * kk + 2 * hi;
      const int kb = ka + 4;
      v2f a00 = *(const v2f*)(arow0 + ka);
      v2f a01 = *(const v2f*)(arow0 + kb);
      v2f a10 = *(const v2f*)(arow1 + ka);
      v2f a11 = *(const v2f*)(arow1 + kb);
      v2f b0 = loadB<PACKED>(w2_l2, 64, ka, col);
      v2f b1 = loadB<PACKED>(w2_l2, 64, kb, col);
      acc0 = wmma4(a00, b0, acc0);
      acc1 = wmma4(a01, b1, acc1);
      acc2 = wmma4(a10, b0, acc2);
      acc3 = wmma4(a11, b1, acc3);
    }
    const v8f accA = acc0 + acc1;
    const v8f accB = acc2 + acc3;
#pragma unroll
    for (int v = 0; v < 8; ++v) {
      out[(size_t)(node0 + v + 8 * hi) * 960 + 640 + col * 5 + i] =
          accA[v] * (attr0[v] * 0.10206207261596575f);
      out[(size_t)(node0 + 16 + v + 8 * hi) * 960 + 640 + col * 5 + i] =
          accB[v] * (attr1[v] * 0.10206207261596575f);
    }
  }
}

// Interleave W[K][C] into k-pair-contiguous form:
// dst[((k>>1)*C + c)*2 + (k&1)] = src[k*C + c]
__global__ void pack_pairs(const float* __restrict__ src, float* __restrict__ dst,
                           int total, int C) {
  const int idx = blockIdx.x * blockDim.x + threadIdx.x;
  if (idx >= total) return;
  const int k = idx / C;
  const int c = idx - k * C;
  dst[((((k >> 1) * C) + c) << 1) + (k & 1)] = src[idx];
}

extern "C" void kernel_launch(void* const* d_in, const int* in_sizes, int n_in,
                              void* d_out, int out_size, void* d_ws, size_t ws_size,
                              hipStream_t stream) {
  (void)n_in; (void)out_size;
  const float* node_input = (const float*)d_in[0];
  const float* node_attr  = (const float*)d_in[1];
  const float* w1_s  = (const float*)d_in[2];
  const float* b1_s  = (const float*)d_in[3];
  const float* w1_l1 = (const float*)d_in[4];
  const float* w1_l2 = (const float*)d_in[5];
  const float* w2_s  = (const float*)d_in[6];
  const float* b2_s  = (const float*)d_in[7];
  const float* w2_l1 = (const float*)d_in[8];
  const float* w2_l2 = (const float*)d_in[9];
  float* out = (float*)d_out;

  const int n_nodes = in_sizes[0] / 960;   // 65536
  dim3 grid(n_nodes / 32), block(256);

  // Packed-weight scratch layout (float offsets)
  const size_t O_W1S  = 0;                         // 256*672
  const size_t O_W2S  = O_W1S + 256 * 672;         // 384*256
  const size_t O_W1L1 = O_W2S + 384 * 256;         // 128*192
  const size_t O_W2L1 = O_W1L1 + 128 * 192;        // 192*128
  const size_t O_W1L2 = O_W2L1 + 192 * 128;        // 64*96
  const size_t O_W2L2 = O_W1L2 + 64 * 96;          // 96*64
  const size_t TOTALF = O_W2L2 + 96 * 64;          // 331776 floats

  if (ws_size >= TOTALF * sizeof(float)) {
    float* ws = (float*)d_ws;
    struct { const float* s; size_t off; int total, C; } jobs[6] = {
      {w1_s,  O_W1S,  256 * 672, 672}, {w2_s,  O_W2S,  384 * 256, 256},
      {w1_l1, O_W1L1, 128 * 192, 192}, {w2_l1, O_W2L1, 192 * 128, 128},
      {w1_l2, O_W1L2, 64 * 96,   96},  {w2_l2, O_W2L2, 96 * 64,   64},
    };
    for (int j = 0; j < 6; ++j)
      pack_pairs<<<(jobs[j].total + 255) / 256, 256, 0, stream>>>(
          jobs[j].s, ws + jobs[j].off, jobs[j].total, jobs[j].C);
    fused_gated_mlp<true><<<grid, block, 0, stream>>>(
        node_input, node_attr, ws + O_W1S, b1_s, ws + O_W1L1, ws + O_W1L2,
        ws + O_W2S, b2_s, ws + O_W2L1, ws + O_W2L2, out);
  } else {
    fused_gated_mlp<false><<<grid, block, 0, stream>>>(
        node_input, node_attr, w1_s, b1_s, w1_l1, w1_l2,
        w2_s, b2_s, w2_l1, w2_l2, out);
  }
}